// ExponentialMovingAverageChange_5385888989365
// MI455X (gfx1250) — compile-verified
//
#include <hip/hip_runtime.h>
#include <hip/hip_bf16.h>
#include <stddef.h>

// ---------------------------------------------------------------------------
// out[n] = sum_t w_t * x[t,n]
//   w_{T-1} = 1.1 ;  w_t = -(0.1)^2 * 0.9^(T-2-t)  for 1 <= t <= T-2
// f32 underflow of 0.9^k (k >~ 980) zeroes all earlier weights, so only the
// last T_PROC rows are streamed (96 MiB instead of 256 MiB; ~4 us at 23.3TB/s).
// Pass 1: WMMA f32 16x16x4 weighted reduction -> per-chunk partials in d_ws.
//         Tight inner loop: 2 b32 loads + few VALU + 1 WMMA per 4 rows,
//         pointer-recurrence addressing (row strides in the 24-bit imm).
// Pass 2: deterministic fixed-order combine of partials -> d_out.
// ---------------------------------------------------------------------------

typedef __attribute__((ext_vector_type(2))) float v2f;
typedef __attribute__((ext_vector_type(8))) float v8f;

#define T_DIM 4096
#define N_DIM 16384
#define EMA_W 0.9f
#define LOG2_W (-0.15200309344504995f) /* log2(0.9) */

#define T_PROC 1536               /* rows actually carrying nonzero f32 weight */
#define ROWS_PER_CHUNK 128
#define CHUNKS (T_PROC / ROWS_PER_CHUNK) /* 12 */
#define WAVES_PER_BLOCK 8         /* 256 threads = 8 wave32 */

__global__ void __launch_bounds__(256)
ema_wmma_partial(const float* __restrict__ x, float* __restrict__ partial) {
    const int lane = threadIdx.x & 31;
    const int wave = threadIdx.x >> 5;
    const int n0   = (blockIdx.x * WAVES_PER_BLOCK + wave) * 16; // 16-col tile
    const int t0   = (T_DIM - T_PROC) + blockIdx.y * ROWS_PER_CHUNK;

    // ISA layout for 32-bit A(16x4)/B(4x16): VGPR v holds K=v in lanes 0-15
    // and K=v+2 in lanes 16-31; lane%16 is the M (A, broadcast) / N (B) index.
    const int col  = n0 + (lane & 15);
    const int koff = (lane >> 4) << 1; // 0 for lanes 0-15, 2 for lanes 16-31

    // One per-lane streaming pointer; all row strides become imm offsets.
    const float* p = x + (size_t)(t0 + koff) * N_DIM + col;

    // w(t + k) = w(t) * (1/W)^k ; lane k-offset folded into a constant mul.
    const float invW  = 1.0f / EMA_W;
    const float lmul  = (koff == 0) ? 1.0f : invW * invW; // (1/W)^koff

    v8f acc = {};

#pragma unroll 4
    for (int t = t0; t < t0 + ROWS_PER_CHUNK; t += 4) {
        v2f b;
        b.x = p[0];        // row t + koff
        b.y = p[N_DIM];    // row t + koff + 1

        // Uniform step weight: one exp2f per 4 rows (underflows to 0 exactly
        // like the f32 reference coefficients for old rows).
        const float wstep = -(1.0f - EMA_W) * (1.0f - EMA_W) *
                            exp2f((float)(T_DIM - 2 - t) * LOG2_W);
        v2f a;
        a.x = wstep * lmul;                 // w(t + koff)        (even row)
        float wy = a.x * invW;              // w(t + koff + 1)    (odd row)
        // Row T-1 (odd) carries weight 1 + (1-W) = 1.1; it can only appear as
        // the a.y element of lanes 16-31 in the final step.
        a.y = (t + koff + 1 == T_DIM - 1) ? (2.0f - EMA_W) : wy;

        // Streaming hint 16 rows ahead (global_prefetch_b8); uniform guard
        // keeps it inside the allocation.
        if (t + 16 < T_DIM - 4) {
            __builtin_prefetch(p + (size_t)16 * N_DIM, 0, 1);
        }

        // D = A(16x4,f32) * B(4x16,f32) + C ; accumulate across the chunk
        acc = __builtin_amdgcn_wmma_f32_16x16x4_f32(
            /*neg_a=*/false, a, /*neg_b=*/false, b,
            /*c_mod=*/(short)0, acc, /*reuse_a=*/false, /*reuse_b=*/false);

        p += (size_t)4 * N_DIM;
    }

    // Every row of D holds the same 16 partial sums (A rows identical).
    // C/D layout: VGPR0, lanes 0-15 = (M=0, N=lane) -> take acc[0].
    if (lane < 16) {
        partial[(size_t)blockIdx.y * N_DIM + n0 + lane] = acc[0];
    }
}

__global__ void __launch_bounds__(256)
ema_combine(const float* __restrict__ partial, float* __restrict__ out) {
    const int n = blockIdx.x * blockDim.x + threadIdx.x;
    float s = 0.0f;
#pragma unroll
    for (int c = 0; c < CHUNKS; ++c) {
        s += partial[(size_t)c * N_DIM + n]; // fixed order -> deterministic
    }
    out[n] = s;
}

extern "C" void kernel_launch(void* const* d_in, const int* in_sizes, int n_in,
                              void* d_out, int out_size, void* d_ws, size_t ws_size,
                              hipStream_t stream) {
    (void)in_sizes; (void)n_in; (void)out_size; (void)ws_size;

    const float* x       = (const float*)d_in[0];
    float*       out     = (float*)d_out;
    float*       partial = (float*)d_ws; // CHUNKS * N_DIM floats = 768 KiB

    dim3 grid1(N_DIM / (16 * WAVES_PER_BLOCK), CHUNKS); // (128, 12)
    ema_wmma_partial<<<grid1, 256, 0, stream>>>(x, partial);

    ema_combine<<<N_DIM / 256, 256, 0, stream>>>(partial, out);
}